// gaussian_model_80582176407668
// MI455X (gfx1250) — compile-verified
//
#include <hip/hip_runtime.h>
#include <hip/hip_bf16.h>

typedef __attribute__((ext_vector_type(2))) float v2f;
typedef __attribute__((ext_vector_type(8))) float v8f;

// ---------------------------------------------------------------------------
// Branchless tiled WMMA f32 GEMM: C[M,N] = op(A) * op(B).
// REQUIRES: M,N multiples of 16, K multiple of 4, all operands zero-padded,
// 8B-aligned buffers. One wave per 16x16 C tile; K consumed 4/step via
// V_WMMA_F32_16X16X4_F32. Trans flags are compile-time so contiguous operand
// fragments load as global_load_b64.
//   A(m,k) = TA ? A[k*lda + m] : A[m*lda + k]
//   B(k,n) = TB ? B[n*ldb + k] : B[k*ldb + n]
// A-frag (ISA 16x4 f32): lane<16 -> K=k0,k0+1 ; lane>=16 -> K=k0+2,k0+3
// B-frag / C-frag: K/M rows striped across lanes within a VGPR.
// ---------------------------------------------------------------------------
template <bool TA, bool TB>
__global__ __launch_bounds__(32) void gemm_wmma_f32(
    const float* __restrict__ A, const float* __restrict__ B, float* __restrict__ C,
    int K, int lda, int ldb, int ldc)
{
    const int tile_n = blockIdx.x * 16;
    const int tile_m = blockIdx.y * 16;
    const int lane   = threadIdx.x;
    const int half   = lane >> 4;     // 0 or 1
    const int l15    = lane & 15;

    const int m = tile_m + l15;       // A-frag row
    const int n = tile_n + l15;       // B-frag col

    // aPtr -> A(m, 2*half), advancing 4 in K per step.
    const float* aPtr = TA ? (A + (size_t)(2 * half) * lda + m)
                           : (A + (size_t)m * lda + 2 * half);
    const float* bPtr = TB ? (B + (size_t)n * ldb + 2 * half)
                           : (B + (size_t)(2 * half) * ldb + n);
    const size_t aStep = TA ? (size_t)4 * lda : (size_t)4;
    const size_t bStep = TB ? (size_t)4 : (size_t)4 * ldb;

    v8f acc = {};
    #pragma unroll 4
    for (int k0 = 0; k0 < K; k0 += 4) {
        v2f a, b;
        if constexpr (TA) { a.x = aPtr[0]; a.y = aPtr[lda]; }   // strided: 2x b32
        else              { a = *(const v2f*)aPtr; }            // contiguous: b64
        if constexpr (TB) { b = *(const v2f*)bPtr; }            // contiguous: b64
        else              { b.x = bPtr[0]; b.y = bPtr[ldb]; }   // strided: 2x b32
        // 8 args: (neg_a, A, neg_b, B, c_mod, C, reuse_a, reuse_b)
        acc = __builtin_amdgcn_wmma_f32_16x16x4_f32(
            false, a, false, b, (short)0, acc, false, false);
        aPtr += aStep;
        bPtr += bStep;
    }

    float* cPtr = C + (size_t)(tile_m + 8 * half) * ldc + n;
    #pragma unroll
    for (int r = 0; r < 8; ++r)
        cPtr[(size_t)r * ldc] = acc[r];
}

// ---------------------------------------------------------------------------
// One-workgroup parallel cyclic Jacobi eigensolver, symmetric n x n (n even).
// A and V in dynamic LDS (n=150: ~181 KB < 320 KB/WGP). Round-robin pairing,
// n/2 disjoint rotations/round, snapshot angles -> row phase -> column phase.
// Eigenvalues -> wout[0..n); eigenvectors -> Vout (leading dim ldv, columns).
// ---------------------------------------------------------------------------
__global__ void jacobi_eigh(const float* __restrict__ Kin,
                            float* __restrict__ Vout, float* __restrict__ wout,
                            int n, int ldv, int sweeps)
{
    extern __shared__ float smem[];
    float* A  = smem;                 // n*n
    float* V  = smem + (size_t)n * n; // n*n
    int*   arr = (int*)(smem + 2 * (size_t)n * n);  // n
    float* cs = (float*)(arr + n);    // n/2
    float* sn = cs + n / 2;           // n/2

    const int tid = threadIdx.x;
    const int nthr = blockDim.x;
    const int nn = n * n;
    const int npairs = n / 2;

    for (int i = tid; i < nn; i += nthr) {
        A[i] = Kin[i];
        V[i] = ((i / n) == (i % n)) ? 1.0f : 0.0f;
    }
    for (int i = tid; i < n; i += nthr) arr[i] = i;
    __syncthreads();

    for (int sweep = 0; sweep < sweeps; ++sweep) {
        for (int round = 0; round < n - 1; ++round) {
            // Phase 1: rotation angles from pre-round snapshot.
            for (int kp = tid; kp < npairs; kp += nthr) {
                int p = arr[kp], q = arr[n - 1 - kp];
                if (p > q) { int t = p; p = q; q = t; }
                const float app = A[p * n + p];
                const float aqq = A[q * n + q];
                const float apq = A[p * n + q];
                float c = 1.0f, s = 0.0f;
                if (fabsf(apq) > 1e-12f) {
                    const float tau = (aqq - app) / (2.0f * apq);
                    const float rt  = sqrtf(1.0f + tau * tau);
                    const float t   = (tau >= 0.0f) ? 1.0f / (tau + rt)
                                                    : 1.0f / (tau - rt);
                    c = 1.0f / sqrtf(1.0f + t * t);
                    s = t * c;
                }
                cs[kp] = c; sn[kp] = s;
            }
            __syncthreads();

            // Phase 2: row rotations A = J^T A (pairs disjoint).
            const int work = npairs * n;
            for (int idx = tid; idx < work; idx += nthr) {
                const int kp = idx / n, j = idx % n;
                int p = arr[kp], q = arr[n - 1 - kp];
                if (p > q) { int t = p; p = q; q = t; }
                const float c = cs[kp], s = sn[kp];
                const float apj = A[p * n + j], aqj = A[q * n + j];
                A[p * n + j] = c * apj - s * aqj;
                A[q * n + j] = s * apj + c * aqj;
            }
            __syncthreads();

            // Phase 3: column rotations A = A J; accumulate V = V J.
            for (int idx = tid; idx < work; idx += nthr) {
                const int kp = idx / n, i = idx % n;
                int p = arr[kp], q = arr[n - 1 - kp];
                if (p > q) { int t = p; p = q; q = t; }
                const float c = cs[kp], s = sn[kp];
                const float aip = A[i * n + p], aiq = A[i * n + q];
                A[i * n + p] = c * aip - s * aiq;
                A[i * n + q] = s * aip + c * aiq;
                const float vip = V[i * n + p], viq = V[i * n + q];
                V[i * n + p] = c * vip - s * viq;
                V[i * n + q] = s * vip + c * viq;
            }
            __syncthreads();

            // Rotate schedule (arr[0] fixed), in parallel.
            int val = 0;
            const bool mv = (tid >= 1 && tid < n);
            if (mv) val = arr[(tid == 1) ? (n - 1) : (tid - 1)];
            __syncthreads();
            if (mv) arr[tid] = val;
            __syncthreads();
        }
    }

    for (int i = tid; i < nn; i += nthr)
        Vout[(size_t)(i / n) * ldv + (i % n)] = V[i];
    for (int i = tid; i < n; i += nthr) wout[i] = A[i * n + i];
}

// ---------------------------------------------------------------------------
// Builders / elementwise kernels (write into pre-zeroed padded buffers).
// ---------------------------------------------------------------------------
__global__ void zero_ws(float* __restrict__ w, int n)
{
    const int idx = blockIdx.x * blockDim.x + threadIdx.x;
    if (idx < n) w[idx] = 0.0f;
}

__global__ void se_spatial(const float* __restrict__ X1, int n1,
                           const float* __restrict__ X2, int n2, int ld,
                           const float* __restrict__ logll,
                           const float* __restrict__ logle,
                           float jitter, float* __restrict__ out)
{
    const int idx = blockIdx.x * blockDim.x + threadIdx.x;
    if (idx >= n1 * n2) return;
    const int i = idx / n2, j = idx % n2;
    const float ll = expf(logll[0]);
    const float le = expf(logle[0]);
    const float dx = X1[i * 3 + 0] - X2[j * 3 + 0];
    const float dy = X1[i * 3 + 1] - X2[j * 3 + 1];
    const float dz = X1[i * 3 + 2] - X2[j * 3 + 2];
    const float d2 = (dx * dx + dy * dy) / (ll * ll) + (dz * dz) / (le * le);
    float v = expf(-d2);
    if (jitter != 0.0f && i == j) v += jitter;
    out[i * ld + j] = v;
}

__global__ void se_temporal(const float* __restrict__ T1, int n1,
                            const float* __restrict__ T2, int n2, int ld,
                            const float* __restrict__ loglt,
                            float jitter, float* __restrict__ out)
{
    const int idx = blockIdx.x * blockDim.x + threadIdx.x;
    if (idx >= n1 * n2) return;
    const int i = idx / n2, j = idx % n2;
    const float lt = expf(loglt[0]);
    const float dt = (T1[i] - T2[j]) / lt;
    float v = expf(-dt * dt);
    if (jitter != 0.0f && i == j) v += jitter;
    out[i * ld + j] = v;
}

// Zpad[k, n] = stData[n, k]  (materialize Z = stData^T into padded buffer)
__global__ void transpose_pad(const float* __restrict__ src, int rows, int cols,
                              float* __restrict__ dst, int ldd)
{
    const int idx = blockIdx.x * blockDim.x + threadIdx.x;
    if (idx >= rows * cols) return;
    const int r = idx / cols, c = idx % cols;     // src[r, c]
    dst[c * ldd + r] = src[r * cols + c];
}

__global__ void build_inv(const float* __restrict__ wt, const float* __restrict__ ws,
                          const float* __restrict__ lognoise,
                          float* __restrict__ out, int Nt, int Ns, int ld)
{
    const int idx = blockIdx.x * blockDim.x + threadIdx.x;
    if (idx >= Nt * Ns) return;
    const int jt = idx / Ns, js = idx % Ns;
    out[jt * ld + js] = 1.0f / (wt[jt] * ws[js] + expf(lognoise[0]));
}

__global__ void elem_mul(const float* __restrict__ a, const float* __restrict__ b,
                         float* __restrict__ out, int n)
{
    const int idx = blockIdx.x * blockDim.x + threadIdx.x;
    if (idx < n) out[idx] = a[idx] * b[idx];
}

// out[0..Ms*Mt)  : yPred[is,it] = P[it,is] ; out[Ms*Mt..) : yVar = exp(sg)-Q[it,is]
__global__ void finalize_out(const float* __restrict__ P, const float* __restrict__ Q,
                             int ldP, const float* __restrict__ logsig,
                             float* __restrict__ out, int Ms, int Mt)
{
    const int idx = blockIdx.x * blockDim.x + threadIdx.x;
    if (idx >= Ms * Mt) return;
    const int is = idx / Mt, it = idx % Mt;
    out[idx]           = P[it * ldP + is];
    out[Ms * Mt + idx] = expf(logsig[0]) - Q[it * ldP + is];
}

// ---------------------------------------------------------------------------
// Host orchestration
// ---------------------------------------------------------------------------
static inline void launch_gemm(const float* A, const float* B, float* C,
                               int M, int N, int K, int lda, int ldb, int ldc,
                               int tA, int tB, hipStream_t stream)
{
    dim3 grid(N / 16, M / 16);
    if (!tA && !tB)
        gemm_wmma_f32<false, false><<<grid, 32, 0, stream>>>(A, B, C, K, lda, ldb, ldc);
    else if (!tA && tB)
        gemm_wmma_f32<false, true ><<<grid, 32, 0, stream>>>(A, B, C, K, lda, ldb, ldc);
    else if (tA && !tB)
        gemm_wmma_f32<true,  false><<<grid, 32, 0, stream>>>(A, B, C, K, lda, ldb, ldc);
    else
        gemm_wmma_f32<true,  true ><<<grid, 32, 0, stream>>>(A, B, C, K, lda, ldb, ldc);
}

extern "C" void kernel_launch(void* const* d_in, const int* in_sizes, int n_in,
                              void* d_out, int out_size, void* d_ws, size_t ws_size,
                              hipStream_t stream)
{
    (void)in_sizes; (void)n_in; (void)out_size; (void)ws_size;
    const int Ns = 150, Nt = 40, Ms = 200, Mt = 30;
    const int NsP = 160, NtP = 48, MsP = 208, MtP = 32;   // tile-padded dims

    const float* spaceC  = (const float*)d_in[0];   // [150,3]
    const float* timeC   = (const float*)d_in[1];   // [40,1]
    const float* stData  = (const float*)d_in[2];   // [150,40]
    const float* tSpaceC = (const float*)d_in[3];   // [200,3]
    const float* tTimeC  = (const float*)d_in[4];   // [30,1]
    const float* logll   = (const float*)d_in[5];
    const float* logle   = (const float*)d_in[6];
    const float* loglt   = (const float*)d_in[7];
    const float* lognz   = (const float*)d_in[8];
    const float* logsg   = (const float*)d_in[9];
    float* out = (float*)d_out;

    float* W = (float*)d_ws;
    size_t off = 0;
    auto alloc = [&](size_t nElem) { float* p = W + off; off += nElem; return p; };

    float* Ks    = alloc((size_t)Ns * Ns);     // Jacobi input (contiguous 150x150)
    float* Kt    = alloc((size_t)Nt * Nt);     // Jacobi input (40x40)
    float* VsE   = alloc((size_t)NsP * NsP);   // eigvecs, ld=NsP, pads zero
    float* wsE   = alloc(NsP);
    float* VtE   = alloc((size_t)NtP * NtP);   // eigvecs, ld=NtP
    float* wtE   = alloc(NtP);
    float* Inv   = alloc((size_t)NtP * NsP);   // valid 40x150, pads zero
    float* Zpad  = alloc((size_t)NtP * NsP);   // stData^T, padded
    float* Zt1   = alloc((size_t)NtP * NsP);
    float* Zt    = alloc((size_t)NtP * NsP);
    float* Bm    = alloc((size_t)NtP * NsP);
    float* A1    = alloc((size_t)NtP * NsP);
    float* Alpha = alloc((size_t)NtP * NsP);
    float* Tt    = alloc((size_t)MtP * NtP);   // valid 30x40
    float* Ts    = alloc((size_t)MsP * NsP);   // valid 200x150
    float* P1    = alloc((size_t)MtP * NsP);
    float* P     = alloc((size_t)MtP * MsP);   // valid 30x200
    float* At    = alloc((size_t)MtP * NtP);
    float* As    = alloc((size_t)MsP * NsP);
    float* At2   = alloc((size_t)MtP * NtP);
    float* As2   = alloc((size_t)MsP * NsP);
    float* Q1    = alloc((size_t)MtP * NsP);
    float* Q     = alloc((size_t)MtP * MsP);

    const int TPB = 256;
    auto blk = [&](int n) { return (n + TPB - 1) / TPB; };

    // 0) Zero workspace so all pads are zero (pads then propagate as zeros).
    zero_ws<<<blk((int)off), TPB, 0, stream>>>(W, (int)off);

    // 1) Training kernels (contiguous, for Jacobi)
    se_spatial<<<blk(Ns * Ns), TPB, 0, stream>>>(spaceC, Ns, spaceC, Ns, Ns,
                                                 logll, logle, 0.1f, Ks);
    se_temporal<<<blk(Nt * Nt), TPB, 0, stream>>>(timeC, Nt, timeC, Nt, Nt,
                                                  loglt, 0.1f, Kt);

    // 2) Eigendecompositions (LDS-resident parallel Jacobi); eigvecs land padded.
    {
        size_t ldsS = (2u * Ns * Ns) * 4 + Ns * 4 + Ns * 4;
        jacobi_eigh<<<1, 256, ldsS, stream>>>(Ks, VsE, wsE, Ns, NsP, 14);
        size_t ldsT = (2u * Nt * Nt) * 4 + Nt * 4 + Nt * 4;
        jacobi_eigh<<<1, 256, ldsT, stream>>>(Kt, VtE, wtE, Nt, NtP, 14);
    }

    // 3) InvMat[jt,js] = 1/(wt*ws + noise)  (valid block only; pads stay 0)
    build_inv<<<blk(Nt * Ns), TPB, 0, stream>>>(wtE, wsE, lognz, Inv, Nt, Ns, NsP);

    // 3b) Zpad = stData^T (padded)
    transpose_pad<<<blk(Ns * Nt), TPB, 0, stream>>>(stData, Ns, Nt, Zpad, NsP);

    // 4) Alpha = Vt ((Vt^T Z Vs) o Inv) Vs^T
    launch_gemm(VtE, Zpad, Zt1, NtP, NsP, NtP, NtP, NsP, NsP, 1, 0, stream);
    launch_gemm(Zt1, VsE, Zt, NtP, NsP, NsP, NsP, NsP, NsP, 0, 0, stream);
    elem_mul<<<blk(NtP * NsP), TPB, 0, stream>>>(Zt, Inv, Bm, NtP * NsP);
    launch_gemm(VtE, Bm, A1, NtP, NsP, NtP, NtP, NsP, NsP, 0, 0, stream);
    launch_gemm(A1, VsE, Alpha, NtP, NsP, NsP, NsP, NsP, NsP, 0, 1, stream);

    // 5) Test cross-kernels (into padded, pre-zeroed buffers)
    se_temporal<<<blk(Mt * Nt), TPB, 0, stream>>>(tTimeC, Mt, timeC, Nt, NtP,
                                                  loglt, 0.0f, Tt);
    se_spatial<<<blk(Ms * Ns), TPB, 0, stream>>>(tSpaceC, Ms, spaceC, Ns, NsP,
                                                 logll, logle, 0.0f, Ts);

    // 6) yPred core: P = Tt Alpha Ts^T
    launch_gemm(Tt, Alpha, P1, MtP, NsP, NtP, NtP, NsP, NsP, 0, 0, stream);
    launch_gemm(P1, Ts, P, MtP, MsP, NsP, NsP, NsP, MsP, 0, 1, stream);

    // 7) yVar core: Q = (At o At) Inv (As o As)^T
    launch_gemm(Tt, VtE, At, MtP, NtP, NtP, NtP, NtP, NtP, 0, 0, stream);
    launch_gemm(Ts, VsE, As, MsP, NsP, NsP, NsP, NsP, NsP, 0, 0, stream);
    elem_mul<<<blk(MtP * NtP), TPB, 0, stream>>>(At, At, At2, MtP * NtP);
    elem_mul<<<blk(MsP * NsP), TPB, 0, stream>>>(As, As, As2, MsP * NsP);
    launch_gemm(At2, Inv, Q1, MtP, NsP, NtP, NtP, NsP, NsP, 0, 0, stream);
    launch_gemm(Q1, As2, Q, MtP, MsP, NsP, NsP, NsP, MsP, 0, 1, stream);

    // 8) Transposed outputs
    finalize_out<<<blk(Ms * Mt), TPB, 0, stream>>>(P, Q, MsP, logsg, out, Ms, Mt);
}